// VQ_VAE_multi_factor_78589311582887
// MI455X (gfx1250) — compile-verified
//
#include <hip/hip_runtime.h>

#define OBS_DIM  32
#define N_CODE   512
#define CODE_DIM 16
#define H        256
#define BATCH    8192
#define ROWS     (BATCH * OBS_DIM)   // 262144 encoder rows
#define RSTRIDE  260                 // padded LDS row stride: (4m+k)%64 -> conflict-free b64 A frags
#define ENC_ROWS_PER_BLOCK 128       // 8 waves x 16 rows
#define KC       64                  // W1 K-chunk staged in LDS
#define NCHUNK   (H / KC)            // 4 chunks

typedef float v2f __attribute__((ext_vector_type(2)));
typedef float v8f __attribute__((ext_vector_type(8)));
typedef int   v4i __attribute__((ext_vector_type(4)));

#if defined(__has_builtin)
#if __has_builtin(__builtin_amdgcn_global_load_async_to_lds_b128) && \
    __has_builtin(__builtin_amdgcn_s_wait_asynccnt)
#define USE_ASYNC 1
#endif
#endif
#ifndef USE_ASYNC
#define USE_ASYNC 0
#endif

// Builtin signature (probe-verified via diagnostic): AS1 v4i* src, AS3 v4i* dst, imm, imm
#define GPTR(p) ((__attribute__((address_space(1))) v4i*)(p))
#define LPTR(p) ((__attribute__((address_space(3))) v4i*)(p))

__device__ __forceinline__ v8f wmma4(v2f a, v2f b, v8f c) {
  // D(16x16,f32) = A(16x4,f32) * B(4x16,f32) + C
  return __builtin_amdgcn_wmma_f32_16x16x4_f32(false, a, false, b, (short)0, c, false, false);
}

__device__ __forceinline__ v8f zero8() {
  v8f z = {0.f, 0.f, 0.f, 0.f, 0.f, 0.f, 0.f, 0.f};
  return z;
}

// ---------------------------------------------------------------------------
// Kernel 1: fused encoder (33->256->256->16, one-hot layer-0 collapsed) +
//           VQ quantize (distance GEMM on WMMA + shfl_xor argmin).
// 256 threads = 8 waves; wave owns a 16-row tile; block owns 128 rows.
// W1 double-buffered in LDS via async-to-LDS: copy of chunk i+1 overlaps
// compute of chunk i; one barrier + one s_wait_asynccnt per chunk.
// ---------------------------------------------------------------------------
__global__ __launch_bounds__(256)
void vqvae_encode_quant(const float* __restrict__ obs,
                        const float* __restrict__ W0, const float* __restrict__ b0,
                        const float* __restrict__ W1, const float* __restrict__ b1,
                        const float* __restrict__ W2, const float* __restrict__ b2,
                        const float* __restrict__ cb, float* __restrict__ qbuf) {
  __shared__ float sh[ENC_ROWS_PER_BLOCK * RSTRIDE]; // activations (wave-private 16-row slices)
  __shared__ float wch[2][KC * H];                    // W1 K-chunks, double-buffered
  __shared__ float scn[N_CODE];                       // |codebook row|^2

  const int tid  = threadIdx.x;
  const int lane = tid & 31;
  const int wave = tid >> 5;
  const int hi   = lane >> 4;          // 0: K={k,k+1}/rows 0-7 ; 1: K={k+2,k+3}/rows 8-15
  const int ln   = lane & 15;
  const int kOff = hi << 1;
  const int rowBase = blockIdx.x * ENC_ROWS_PER_BLOCK;
  const int rb      = wave * 16;

  // stage one 64x256 W1 chunk (64KB) into wch[buf]; 16 x b128 per thread
  auto stage = [&](int buf, int kc) {
#pragma unroll
    for (int j = 0; j < (KC * H) / (256 * 4); ++j) {
      int f = tid + j * 256;                           // float4 index
#if USE_ASYNC
      __builtin_amdgcn_global_load_async_to_lds_b128(
          GPTR(W1 + kc * H + f * 4), LPTR(&wch[buf][f * 4]), 0, 0);
#else
      float4 v = *(const float4*)(W1 + kc * H + f * 4);
      *(float4*)(&wch[buf][f * 4]) = v;
#endif
    }
  };

  // codebook norms
  for (int j = tid; j < N_CODE; j += 256) {
    float s = 0.f;
#pragma unroll
    for (int k = 0; k < CODE_DIM; ++k) { float c = cb[j * CODE_DIM + k]; s += c * c; }
    scn[j] = s;
  }

  stage(0, 0);   // prefetch first W1 chunk while Phase A runs

  // Phase A: h0 = relu(x * W0[0,:] + W0[1+i,:] + b0)  (one-hot => no GEMM)
  for (int t = tid; t < ENC_ROWS_PER_BLOCK * H; t += 256) {
    int rl = t >> 8;
    int c  = t & (H - 1);
    int g  = rowBase + rl;
    int bi = g >> 5, i = g & 31;
    float x = obs[bi * OBS_DIM + i];
    float h = fmaf(x, W0[c], W0[(1 + i) * H + c] + b0[c]);
    sh[rl * RSTRIDE + c] = fmaxf(h, 0.f);
  }

  // Phase B: h1 = relu(h0 @ W1 + b1)  M=16 N=256 K=256 per wave, W1 double-buffered
  v8f acc[16];
#pragma unroll
  for (int nt = 0; nt < 16; ++nt) acc[nt] = zero8();

  for (int ic = 0; ic < NCHUNK; ++ic) {
    const int kc  = ic * KC;
    const int cur = ic & 1;
#if USE_ASYNC
    __builtin_amdgcn_s_wait_asynccnt(0);   // our share of chunk ic is in LDS
#endif
    __syncthreads();                       // everyone's share in; prev compute done
    if (ic + 1 < NCHUNK) stage(1 - cur, kc + KC);   // overlap next copy with compute

    for (int k2 = 0; k2 < KC; k2 += 4) {
      v2f a = *(const v2f*)&sh[(rb + ln) * RSTRIDE + kc + k2 + kOff]; // ds_load_b64
      int kb = k2 + kOff;
#pragma unroll
      for (int nt = 0; nt < 16; ++nt) {
        int n = nt * 16 + ln;
        v2f b = { wch[cur][kb * H + n], wch[cur][(kb + 1) * H + n] }; // conflict-free LDS
        acc[nt] = wmma4(a, b, acc[nt]);
      }
    }
  }
  __syncthreads();   // last chunk consumed everywhere (keeps buffers consistent)

#pragma unroll
  for (int nt = 0; nt < 16; ++nt) {
    int n = nt * 16 + ln;
    float bias = b1[n];
#pragma unroll
    for (int r = 0; r < 8; ++r)
      sh[(rb + r + hi * 8) * RSTRIDE + n] = fmaxf(acc[nt][r] + bias, 0.f);
  }

  // Phase C: emb = h1 @ W2 + b2  M=16 N=16 K=256 (W2 is 16KB, WGP$-resident)
  v8f e = zero8();
  for (int k = 0; k < H; k += 4) {
    v2f a = *(const v2f*)&sh[(rb + ln) * RSTRIDE + k + kOff];
    int kb = k + kOff;
    v2f b = { W2[kb * CODE_DIM + ln], W2[(kb + 1) * CODE_DIM + ln] };
    e = wmma4(a, b, e);
  }
  {
    float bias = b2[ln];
#pragma unroll
    for (int r = 0; r < 8; ++r) e[r] += bias;
  }
#pragma unroll
  for (int r = 0; r < 8; ++r)
    sh[(rb + r + hi * 8) * RSTRIDE + ln] = e[r];   // stash emb (h1 fully consumed)

  // Phase D: scores = emb @ cb^T  M=16 N=512 K=16 ; d = |c|^2 - 2*score
  v2f a4[4];
#pragma unroll
  for (int kk = 0; kk < 4; ++kk)
    a4[kk] = *(const v2f*)&sh[(rb + ln) * RSTRIDE + 4 * kk + kOff];

  float bestv[8];
  int   besti[8];
#pragma unroll
  for (int r = 0; r < 8; ++r) { bestv[r] = 3.4e38f; besti[r] = 0; }

  for (int nt = 0; nt < 32; ++nt) {
    int n = nt * 16 + ln;
    v8f s = zero8();
#pragma unroll
    for (int kk = 0; kk < 4; ++kk) {
      v2f b = *(const v2f*)&cb[n * CODE_DIM + 4 * kk + kOff];   // global_load_b64
      s = wmma4(a4[kk], b, s);
    }
    float cn = scn[n];
#pragma unroll
    for (int r = 0; r < 8; ++r) {
      float d = cn - 2.f * s[r];
      bool better = (d < bestv[r]) || (d == bestv[r] && n < besti[r]);
      bestv[r] = better ? d : bestv[r];
      besti[r] = better ? n : besti[r];
    }
  }
#pragma unroll
  for (int r = 0; r < 8; ++r) {
    float v = bestv[r];
    int  ix = besti[r];
#pragma unroll
    for (int off = 1; off < 16; off <<= 1) {
      float ov = __shfl_xor(v, off, 32);
      int   oi = __shfl_xor(ix, off, 32);
      bool take = (ov < v) || (ov == v && oi < ix);
      v  = take ? ov : v;
      ix = take ? oi : ix;
    }
    int g  = rowBase + rb + r + hi * 8;
    int bi = g >> 5, i = g & 31;
    qbuf[bi * (OBS_DIM * CODE_DIM) + i * CODE_DIM + ln] = cb[ix * CODE_DIM + ln];
  }
}

// ---------------------------------------------------------------------------
// Kernel 2: decoder 512->256->256->32 on 8192 rows (weights ~0.8MB, L2-hot;
// only 128 blocks so direct global B reads are cheap here)
// ---------------------------------------------------------------------------
__global__ __launch_bounds__(128)
void vqvae_decode(const float* __restrict__ qbuf,
                  const float* __restrict__ W0, const float* __restrict__ b0,
                  const float* __restrict__ W1, const float* __restrict__ b1,
                  const float* __restrict__ W2, const float* __restrict__ b2,
                  float* __restrict__ out) {
  __shared__ float sh[64 * RSTRIDE];

  const int tid  = threadIdx.x;
  const int lane = tid & 31;
  const int wave = tid >> 5;
  const int hi   = lane >> 4;
  const int ln   = lane & 15;
  const int kOff = hi << 1;
  const int rowBase = blockIdx.x * 64;
  const int rb      = wave * 16;

  // Layer 0: h = relu(q @ W0 + b0)  M=16 N=256 K=512 ; A frags direct from L2 (b64)
  v8f acc[16];
#pragma unroll
  for (int nt = 0; nt < 16; ++nt) acc[nt] = zero8();
  const float* aRow = &qbuf[(rowBase + rb + ln) * (OBS_DIM * CODE_DIM)];
  for (int k = 0; k < OBS_DIM * CODE_DIM; k += 4) {
    v2f a = *(const v2f*)&aRow[k + kOff];
#pragma unroll
    for (int nt = 0; nt < 16; ++nt) {
      int n  = nt * 16 + ln;
      int kb = k + kOff;
      v2f b = { W0[kb * H + n], W0[(kb + 1) * H + n] };
      acc[nt] = wmma4(a, b, acc[nt]);
    }
  }
#pragma unroll
  for (int nt = 0; nt < 16; ++nt) {
    int n = nt * 16 + ln;
    float bias = b0[n];
#pragma unroll
    for (int r = 0; r < 8; ++r)
      sh[(rb + r + hi * 8) * RSTRIDE + n] = fmaxf(acc[nt][r] + bias, 0.f);
  }

  // Layer 1: 256 -> 256
#pragma unroll
  for (int nt = 0; nt < 16; ++nt) acc[nt] = zero8();
  for (int k = 0; k < H; k += 4) {
    v2f a = *(const v2f*)&sh[(rb + ln) * RSTRIDE + k + kOff];
#pragma unroll
    for (int nt = 0; nt < 16; ++nt) {
      int n  = nt * 16 + ln;
      int kb = k + kOff;
      v2f b = { W1[kb * H + n], W1[(kb + 1) * H + n] };
      acc[nt] = wmma4(a, b, acc[nt]);
    }
  }
#pragma unroll
  for (int nt = 0; nt < 16; ++nt) {
    int n = nt * 16 + ln;
    float bias = b1[n];
#pragma unroll
    for (int r = 0; r < 8; ++r)
      sh[(rb + r + hi * 8) * RSTRIDE + n] = fmaxf(acc[nt][r] + bias, 0.f);
  }

  // Layer 2: 256 -> 32 (two n-tiles), emit recon
  v8f o0 = zero8(), o1 = zero8();
  for (int k = 0; k < H; k += 4) {
    v2f a = *(const v2f*)&sh[(rb + ln) * RSTRIDE + k + kOff];
    int kb = k + kOff;
    v2f bv0 = { W2[kb * OBS_DIM + ln],      W2[(kb + 1) * OBS_DIM + ln] };
    v2f bv1 = { W2[kb * OBS_DIM + 16 + ln], W2[(kb + 1) * OBS_DIM + 16 + ln] };
    o0 = wmma4(a, bv0, o0);
    o1 = wmma4(a, bv1, o1);
  }
  float bo0 = b2[ln], bo1 = b2[16 + ln];
#pragma unroll
  for (int r = 0; r < 8; ++r) {
    int g = rowBase + rb + r + hi * 8;
    out[g * OBS_DIM + ln]      = o0[r] + bo0;
    out[g * OBS_DIM + 16 + ln] = o1[r] + bo1;
  }
}

extern "C" void kernel_launch(void* const* d_in, const int* in_sizes, int n_in,
                              void* d_out, int out_size, void* d_ws, size_t ws_size,
                              hipStream_t stream) {
  (void)in_sizes; (void)n_in; (void)out_size; (void)ws_size;
  const float* obs = (const float*)d_in[0];
  const float* eW0 = (const float*)d_in[1];
  const float* eb0 = (const float*)d_in[2];
  const float* eW1 = (const float*)d_in[3];
  const float* eb1 = (const float*)d_in[4];
  const float* eW2 = (const float*)d_in[5];
  const float* eb2 = (const float*)d_in[6];
  const float* dW0 = (const float*)d_in[7];
  const float* db0 = (const float*)d_in[8];
  const float* dW1 = (const float*)d_in[9];
  const float* db1 = (const float*)d_in[10];
  const float* dW2 = (const float*)d_in[11];
  const float* db2 = (const float*)d_in[12];
  const float* cb  = (const float*)d_in[13];

  float* qbuf = (float*)d_ws;        // 8192*512 floats = 16 MB scratch (L2-resident)
  float* out  = (float*)d_out;

  vqvae_encode_quant<<<ROWS / ENC_ROWS_PER_BLOCK, 256, 0, stream>>>(
      obs, eW0, eb0, eW1, eb1, eW2, eb2, cb, qbuf);
  vqvae_decode<<<BATCH / 64, 128, 0, stream>>>(qbuf, dW0, db0, dW1, db1, dW2, db2, out);
}